// DynamicsModel_53532472377977
// MI455X (gfx1250) — compile-verified
//
#include <hip/hip_runtime.h>

typedef float v2f __attribute__((ext_vector_type(2)));
typedef float v8f __attribute__((ext_vector_type(8)));

#define T_STEPS 512
// Workspace layout (floats):
//   [0, 4096)     BfragX: 64 x-tiles x 32 lanes x 2 (v2f per lane, lanes>=16 zero)
//   [4096, 6144)  BfragU: 32 u-tiles x 32 lanes x 2
#define WS_FRAG_FLOATS 6144
#define WS_U_OFF 4096

// ---------------------------------------------------------------------------
// Kernel 1: build per-timestep B fragments from powers of M = A - B K,
// emitted directly in the v_wmma_f32_16x16x4_f32 B-operand per-lane layout:
//   lane l < 16 : { Q[0][16n+l], Q[1][16n+l] }   (K=0,1 rows)
//   lane l >= 16: { 0, 0 }                        (K=2,3 zero pad)
// where Qx[k][2t+d] = (M^t)[d][k], Qu[k][t] = -(K0*(M^t)[0][k]+K1*(M^t)[1][k])
// ---------------------------------------------------------------------------
__global__ __launch_bounds__(T_STEPS) void build_powers(
    const float* __restrict__ A, const float* __restrict__ B,
    const float* __restrict__ K, float* __restrict__ ws) {
  __shared__ float Msq[9][4];  // M^(2^j), row-major 2x2
  const int t = threadIdx.x;
  if (t == 0) {
    const float K0 = K[0], K1 = K[1];
    // M = A - B@K (B is 2x1, K is 1x2)
    Msq[0][0] = A[0] - B[0] * K0;
    Msq[0][1] = A[1] - B[0] * K1;
    Msq[0][2] = A[2] - B[1] * K0;
    Msq[0][3] = A[3] - B[1] * K1;
    for (int j = 1; j < 9; ++j) {
      const float a00 = Msq[j-1][0], a01 = Msq[j-1][1];
      const float a10 = Msq[j-1][2], a11 = Msq[j-1][3];
      Msq[j][0] = a00 * a00 + a01 * a10;
      Msq[j][1] = a00 * a01 + a01 * a11;
      Msq[j][2] = a10 * a00 + a11 * a10;
      Msq[j][3] = a10 * a01 + a11 * a11;
    }
  }
  __syncthreads();

  // Zero the lanes>=16 (K-pad) fragment slots: flat index i has lane = (i/2)%32,
  // so lane>=16  <=>  (i & 32) != 0.  (Real-value writes below all have i&32==0.)
  for (int i = t; i < WS_FRAG_FLOATS; i += T_STEPS)
    if (i & 32) ws[i] = 0.0f;

  // P = M^t via binary exponentiation (powers of M commute).
  float p00 = 1.0f, p01 = 0.0f, p10 = 0.0f, p11 = 1.0f;
  #pragma unroll
  for (int j = 0; j < 9; ++j) {
    if ((t >> j) & 1) {
      const float q00 = Msq[j][0], q01 = Msq[j][1];
      const float q10 = Msq[j][2], q11 = Msq[j][3];
      const float n00 = p00 * q00 + p01 * q10;
      const float n01 = p00 * q01 + p01 * q11;
      const float n10 = p10 * q00 + p11 * q10;
      const float n11 = p10 * q01 + p11 * q11;
      p00 = n00; p01 = n01; p10 = n10; p11 = n11;
    }
  }

  // x_pred columns: col = 2t+d holds (M^t)[d][k] for k=0,1.
  {
    const int c0 = 2 * t + 0;                   // d = 0
    float* e0 = ws + (c0 >> 4) * 64 + (c0 & 15) * 2;
    e0[0] = p00; e0[1] = p01;
    const int c1 = 2 * t + 1;                   // d = 1
    float* e1 = ws + (c1 >> 4) * 64 + (c1 & 15) * 2;
    e1[0] = p10; e1[1] = p11;
  }
  // u_pred column t: Qu[k][t] = -(K0*P[0][k] + K1*P[1][k])
  {
    const float K0 = K[0], K1 = K[1];
    float* eu = ws + WS_U_OFF + (t >> 4) * 64 + (t & 15) * 2;
    eu[0] = -(K0 * p00 + K1 * p10);
    eu[1] = -(K0 * p01 + K1 * p11);
  }
}

// ---------------------------------------------------------------------------
// Kernel 2: Out = X0 (batch x 2, K-padded to 4) @ Q via v_wmma_f32_16x16x4_f32.
// One wave: 16 batch rows, 128 x-cols (8 tiles) + 64 u-cols (4 tiles).
// B fragments arrive pre-laid-out and pre-padded -> one b64 load per tile.
// ---------------------------------------------------------------------------
__global__ __launch_bounds__(256) void wmma_rollout(
    const float* __restrict__ x0, const float* __restrict__ ws,
    float* __restrict__ xout, float* __restrict__ uout, int batch) {
  const int lane  = threadIdx.x & 31;
  const int wave  = (blockIdx.x * blockDim.x + threadIdx.x) >> 5;
  const int NCHUNK = 8;  // 8 chunks of (128 x-cols + 64 u-cols)
  const int b_tile = wave / NCHUNK;
  const int chunk  = wave % NCHUNK;
  if (b_tile * 16 >= batch) return;

  const v2f* __restrict__ BfX = (const v2f*)ws;              // [64][32]
  const v2f* __restrict__ BfU = (const v2f*)(ws + WS_U_OFF); // [32][32]

  // Prefetch all 12 B fragments (unconditional b64 loads, clause-friendly).
  v2f bx[8], bu[4];
  #pragma unroll
  for (int jt = 0; jt < 8; ++jt) bx[jt] = BfX[(chunk * 8 + jt) * 32 + lane];
  #pragma unroll
  for (int jt = 0; jt < 4; ++jt) bu[jt] = BfU[(chunk * 4 + jt) * 32 + lane];

  // A fragment, 16x4 f32: lanes 0-15 hold (M=lane, K=0..1); lanes 16-31 are the
  // zero K=2..3 pad. Load unconditionally (lanes>=16 mirror lanes 0-15, always
  // in bounds), then mask by register select (v_cndmask, no exec branch).
  const int arow = b_tile * 16 + (lane & 15);
  const v2f xa = *(const v2f*)(x0 + 2 * arow);
  v2f a;
  a.x = (lane < 16) ? xa.x : 0.0f;
  a.y = (lane < 16) ? xa.y : 0.0f;

  // D tile layout: VGPR v, lanes 0-15 -> row v, lanes 16-31 -> row v+8; col=lane%16
  const int rbase = b_tile * 16 + ((lane >> 4) << 3);
  const int csub  = lane & 15;

  // ---- x_pred: 8 tiles of 16 columns ----
  #pragma unroll
  for (int jt = 0; jt < 8; ++jt) {
    const int n0 = chunk * 128 + jt * 16;
    v8f c = {};
    c = __builtin_amdgcn_wmma_f32_16x16x4_f32(
        /*neg_a=*/false, a, /*neg_b=*/false, bx[jt],
        /*c_mod=*/(short)0, c, /*reuse_a=*/false, /*reuse_b=*/false);
    float* p = xout + (size_t)rbase * (2 * T_STEPS) + n0 + csub;
    #pragma unroll
    for (int v = 0; v < 8; ++v)
      __builtin_nontemporal_store(c[v], p + (size_t)v * (2 * T_STEPS));
  }

  // ---- u_pred: 4 tiles of 16 columns ----
  #pragma unroll
  for (int jt = 0; jt < 4; ++jt) {
    const int n0 = chunk * 64 + jt * 16;
    v8f c = {};
    c = __builtin_amdgcn_wmma_f32_16x16x4_f32(
        false, a, false, bu[jt], (short)0, c, false, false);
    float* p = uout + (size_t)rbase * T_STEPS + n0 + csub;
    #pragma unroll
    for (int v = 0; v < 8; ++v)
      __builtin_nontemporal_store(c[v], p + (size_t)v * T_STEPS);
  }
}

// ---------------------------------------------------------------------------
extern "C" void kernel_launch(void* const* d_in, const int* in_sizes, int n_in,
                              void* d_out, int out_size, void* d_ws, size_t ws_size,
                              hipStream_t stream) {
  const float* x0 = (const float*)d_in[0];  // (batch, 2)
  const float* A  = (const float*)d_in[1];  // (2, 2)
  const float* B  = (const float*)d_in[2];  // (2, 1)
  const float* K  = (const float*)d_in[3];  // (1, 2)
  const int batch = in_sizes[0] / 2;        // 65536

  float* ws   = (float*)d_ws;                            // fragment tables (24 KB)
  float* xout = (float*)d_out;                           // (batch, 512, 2)
  float* uout = xout + (size_t)batch * T_STEPS * 2;      // (batch, 512, 1)

  build_powers<<<1, T_STEPS, 0, stream>>>(A, B, K, ws);

  const int waves  = ((batch + 15) / 16) * 8;
  const int blocks = (waves * 32 + 255) / 256;
  wmma_rollout<<<blocks, 256, 0, stream>>>(x0, ws, xout, uout, batch);
}